// TransitionDown_661424963759
// MI455X (gfx1250) — compile-verified
//
#include <hip/hip_runtime.h>
#include <hip/hip_bf16.h>

typedef __attribute__((ext_vector_type(2))) float v2f;
typedef __attribute__((ext_vector_type(8))) float v8f;

namespace {
constexpr int kB    = 4;
constexpr int kN    = 8192;
constexpr int kM    = 2048;
constexpr int kK    = 16;     // kNN neighbors
constexpr int kCin  = 64;
constexpr int kCout = 128;
constexpr int kRows = kB * kN;          // 32768 total points
constexpr int kQ    = kB * kM;          // 8192 queries
constexpr int kRowTiles = kRows / 16;   // 2048 GEMM row tiles
constexpr int kPad  = 68;               // LDS row stride (floats): 272B, 16B-aligned,
                                        // bank = (4*lane + k) % 64 -> conflict-free
constexpr float kEps = 1e-5f;
constexpr float kInf = 3.4e38f;

// workspace layout (in floats)
constexpr size_t WS_H     = 0;                                   // h: kRows*kCout
constexpr size_t WS_PS    = WS_H  + (size_t)kRows * kCout;       // partial sums
constexpr size_t WS_PQ    = WS_PS + (size_t)kRowTiles * kCout;   // partial sumsq
constexpr size_t WS_SCALE = WS_PQ + (size_t)kRowTiles * kCout;   // scale: kCout
constexpr size_t WS_SHIFT = WS_SCALE + kCout;                    // shift: kCout
constexpr size_t WS_NBR   = WS_SHIFT + kCout;                    // int region: kQ*kK
} // namespace

// ---------------------------------------------------------------------------
// CDNA5 async copy helpers: GLOBAL_LOAD_ASYNC_TO_LDS_B128 (ASYNCcnt-tracked,
// direct global->LDS, no VGPR round trip). The low 32 bits of a flat pointer
// to __shared__ data are the LDS byte offset (ISA 10.2).
// ---------------------------------------------------------------------------
__device__ __forceinline__ void async_copy_b128(void* lds_dst, const void* gsrc) {
  unsigned lds_off = (unsigned)(size_t)lds_dst;
  asm volatile("global_load_async_to_lds_b128 %0, %1, off"
               :: "v"(lds_off), "v"(gsrc) : "memory");
}
__device__ __forceinline__ void wait_async_zero() {
  asm volatile("s_wait_asynccnt 0x0" ::: "memory");
}

// ---------------------------------------------------------------------------
// Kernel 1: h = x @ W^T via f32 WMMA (16x16x4), plus per-block BN partial stats
// block = 256 threads (8 waves). Block owns 16 rows x 128 cols; wave w owns
// the 16-col slice [16w, 16w+16). A/W tiles staged via async-to-LDS.
// ---------------------------------------------------------------------------
__global__ __launch_bounds__(256) void k_gemm_wmma(
    const float* __restrict__ x, const float* __restrict__ Wm,
    float* __restrict__ h, float* __restrict__ partS, float* __restrict__ partQ) {
  __shared__ alignas(16) float As[16 * kPad];    // 16 rows x 64 (+pad)
  __shared__ alignas(16) float Ws[128 * kPad];   // 128 out-chans x 64 (+pad)

  const int t = threadIdx.x;
  const int rowBase = blockIdx.x * 16;

  // stage A tile: 16 rows * 256B = 256 x 16B chunks, one per thread
  {
    int r = t >> 4, c16 = t & 15;
    async_copy_b128(&As[r * kPad] + c16 * 4,
                    (const char*)x + (size_t)(rowBase + r) * 256 + c16 * 16);
  }
  // stage W: 128 rows * 256B = 2048 x 16B chunks, 8 per thread
  for (int i = t; i < 2048; i += 256) {
    int o = i >> 4, c16 = i & 15;
    async_copy_b128(&Ws[o * kPad] + c16 * 4,
                    (const char*)Wm + (size_t)o * 256 + c16 * 16);
  }
  wait_async_zero();
  __syncthreads();

  const int wave = t >> 5;
  const int lane = t & 31;
  const int half = lane >> 4;   // 0: K={0,1}, 1: K={2,3} per WMMA f32 A/B layout
  const int l15  = lane & 15;
  const int colBase = wave * 16;

  const float* arow = &As[l15 * kPad + half * 2];
  const float* brow = &Ws[(colBase + l15) * kPad + half * 2];

  v8f acc = {};
#pragma unroll
  for (int k = 0; k < kCin; k += 4) {
    v2f a; a.x = arow[k]; a.y = arow[k + 1];
    v2f b; b.x = brow[k]; b.y = brow[k + 1];
    // D = A(16x4,f32) * B(4x16,f32) + C(16x16,f32)
    acc = __builtin_amdgcn_wmma_f32_16x16x4_f32(
        false, a, false, b, (short)0, acc, false, false);
  }

  // write tile + accumulate BN partial stats (deterministic: per-block slots)
  const int chan = colBase + l15;
  float s = 0.f, sq = 0.f;
#pragma unroll
  for (int i = 0; i < 8; ++i) {
    float v = acc[i];
    int row = rowBase + i + half * 8;   // C/D layout: lanes 16-31 hold M=8..15
    h[(size_t)row * kCout + chan] = v;
    s += v;
    sq += v * v;
  }
  s  += __shfl_xor(s, 16);
  sq += __shfl_xor(sq, 16);
  if (half == 0) {
    partS[(size_t)blockIdx.x * kCout + chan] = s;
    partQ[(size_t)blockIdx.x * kCout + chan] = sq;
  }
}

// ---------------------------------------------------------------------------
// Kernel 2: reduce partials -> per-channel scale/shift (deterministic order)
// ---------------------------------------------------------------------------
__global__ __launch_bounds__(kCout) void k_bn_finalize(
    const float* __restrict__ partS, const float* __restrict__ partQ,
    const float* __restrict__ gamma, const float* __restrict__ beta,
    float* __restrict__ scale, float* __restrict__ shift) {
  const int c = threadIdx.x;
  float s = 0.f, q = 0.f;
  for (int i = 0; i < kRowTiles; ++i) {
    s += partS[(size_t)i * kCout + c];
    q += partQ[(size_t)i * kCout + c];
  }
  const float invn = 1.0f / (float)kRows;
  const float mean = s * invn;
  const float var  = q * invn - mean * mean;
  const float sc   = gamma[c] * rsqrtf(var + kEps);
  scale[c] = sc;
  shift[c] = beta[c] - mean * sc;
}

// ---------------------------------------------------------------------------
// Kernel 3: kNN. One wave per query; point cloud tiled through LDS via
// async-to-LDS copies. Per-lane branchless sorted top-16 in registers, then
// cooperative 16-round min-extraction merge across lanes in LDS.
// block = 256 threads = 8 queries; all 8 queries share a batch index.
// ---------------------------------------------------------------------------
__global__ __launch_bounds__(256) void k_knn(
    const float* __restrict__ p, const int* __restrict__ sidx,
    float* __restrict__ p_out, int* __restrict__ nbr) {
  constexpr int TILE = 512;                       // 512 pts * 12B = 6144B = 384 x 16B
  __shared__ alignas(16) float ptile[TILE * 3];
  __shared__ float md[8 * 512];
  __shared__ int   mi[8 * 512];

  const int t = threadIdx.x, wave = t >> 5, lane = t & 31;
  const int q = blockIdx.x * 8 + wave;   // global query id
  const int b = q / kM;                  // uniform across the block (kM % 8 == 0)
  const int m = q - b * kM;

  const int n0 = sidx[b * kM + m];
  const size_t pq = ((size_t)(b * kN + n0)) * 3;
  const float qx = p[pq + 0], qy = p[pq + 1], qz = p[pq + 2];
  if (lane < 3) p_out[(size_t)q * 3 + lane] = (lane == 0) ? qx : (lane == 1) ? qy : qz;

  float bd[kK]; int bi[kK];
#pragma unroll
  for (int i = 0; i < kK; ++i) { bd[i] = kInf; bi[i] = 0; }

  for (int tile = 0; tile < kN; tile += TILE) {
    __syncthreads();   // previous iteration's readers done before overwrite
    {
      const char* gbase = (const char*)p + ((size_t)(b * kN + tile)) * 12;
      async_copy_b128((char*)ptile + t * 16, gbase + t * 16);
      if (t < 128)
        async_copy_b128((char*)ptile + (256 + t) * 16, gbase + (256 + t) * 16);
      wait_async_zero();
    }
    __syncthreads();

#pragma unroll 4
    for (int j = lane; j < TILE; j += 32) {
      float dx = ptile[j * 3 + 0] - qx;
      float dy = ptile[j * 3 + 1] - qy;
      float dz = ptile[j * 3 + 2] - qz;
      float d = dx * dx + dy * dy + dz * dz;
      if (d < bd[kK - 1]) {           // branchless insertion into sorted list
        float cd = d; int ci = tile + j;
#pragma unroll
        for (int s2 = 0; s2 < kK; ++s2) {
          bool sm = cd < bd[s2];
          float td = bd[s2]; int ti = bi[s2];
          bd[s2] = sm ? cd : td;  bi[s2] = sm ? ci : ti;
          cd     = sm ? td : cd;  ci     = sm ? ti : ci;
        }
      }
    }
  }

  // dump per-lane candidates, merge via 16 rounds of cooperative min-extract
  float* wd = &md[wave * 512];
  int*   wi = &mi[wave * 512];
#pragma unroll
  for (int i = 0; i < kK; ++i) { wd[lane * kK + i] = bd[i]; wi[lane * kK + i] = bi[i]; }
  __syncthreads();

  for (int r = 0; r < kK; ++r) {
    float best = kInf; int bpos = 0;
    for (int i = lane; i < 512; i += 32) {
      float v = wd[i];
      if (v < best) { best = v; bpos = i; }
    }
#pragma unroll
    for (int off = 16; off > 0; off >>= 1) {
      float ov = __shfl_xor(best, off);
      int   op = __shfl_xor(bpos, off);
      if (ov < best) { best = ov; bpos = op; }
    }
    if (lane == 0) {
      nbr[(size_t)q * kK + r] = wi[bpos];
      wd[bpos] = kInf;
    }
    __syncthreads();
  }
}

// ---------------------------------------------------------------------------
// Kernel 4: gather neighbor features, apply BN affine + ReLU, max-pool over K.
// block = 128 threads (one per channel), grid = one block per query.
// ---------------------------------------------------------------------------
__global__ __launch_bounds__(kCout) void k_gather_max(
    const float* __restrict__ h, const int* __restrict__ nbr,
    const float* __restrict__ scale, const float* __restrict__ shift,
    float* __restrict__ y) {
  const int q = blockIdx.x;
  const int c = threadIdx.x;
  const int b = q / kM;

  int nidx[kK];
#pragma unroll
  for (int k = 0; k < kK; ++k) {
    nidx[k] = nbr[(size_t)q * kK + k];                 // uniform -> scalar load
    __builtin_prefetch(&h[((size_t)(b * kN + nidx[k])) * kCout + c], 0, 0);
  }

  const float sc = scale[c], sh = shift[c];
  float best = -kInf;
#pragma unroll
  for (int k = 0; k < kK; ++k) {
    float v = h[((size_t)(b * kN + nidx[k])) * kCout + c] * sc + sh;
    best = fmaxf(best, v);
  }
  y[(size_t)q * kCout + c] = fmaxf(best, 0.0f);        // ReLU commutes with max
}

// ---------------------------------------------------------------------------
extern "C" void kernel_launch(void* const* d_in, const int* in_sizes, int n_in,
                              void* d_out, int out_size, void* d_ws, size_t ws_size,
                              hipStream_t stream) {
  (void)in_sizes; (void)n_in; (void)out_size; (void)ws_size;

  const float* x     = (const float*)d_in[0];   // (B,N,Cin)
  const float* p     = (const float*)d_in[1];   // (B,N,3)
  const int*   sidx  = (const int*)d_in[2];     // (B,M)
  const float* Wm    = (const float*)d_in[3];   // (Cout,Cin)
  const float* gamma = (const float*)d_in[4];   // (Cout)
  const float* beta  = (const float*)d_in[5];   // (Cout)

  float* out   = (float*)d_out;
  float* y     = out;                              // (B,M,Cout)
  float* p_out = out + (size_t)kQ * kCout;         // (B,M,3)

  float* ws    = (float*)d_ws;
  float* h     = ws + WS_H;
  float* partS = ws + WS_PS;
  float* partQ = ws + WS_PQ;
  float* scale = ws + WS_SCALE;
  float* shift = ws + WS_SHIFT;
  int*   nbr   = (int*)(ws + WS_NBR);

  k_gemm_wmma<<<kRowTiles, 256, 0, stream>>>(x, Wm, h, partS, partQ);
  k_bn_finalize<<<1, kCout, 0, stream>>>(partS, partQ, gamma, beta, scale, shift);
  k_knn<<<kQ / 8, 256, 0, stream>>>(p, sidx, p_out, nbr);
  k_gather_max<<<kQ, kCout, 0, stream>>>(h, nbr, scale, shift, y);
}